// UNetSphericalTestMax_27015344292189
// MI455X (gfx1250) — compile-verified
//
#include <hip/hip_runtime.h>

// Spherical UNet (Chebyshev graph convs) for MI455X / gfx1250.
// - L*x done as a 4-point stencil (gathering the known nonzeros of the dense L input)
// - cheb GEMMs via V_WMMA_F32_16X16X4_F32 (fp32 matrix pipe, wave32)
//   * weights packed in fragment order so every A/B fragment is one global_load_b64
// - deterministic two-stage BatchNorm stats + fused BN/ReLU
// - 2x2 pool/unpool with argmax scatter

#define NB 128
#define NCHUNK 128  // row chunks for BN partial sums (all M are multiples of 128)

typedef float v2f __attribute__((ext_vector_type(2)));
typedef float v8f __attribute__((ext_vector_type(8)));

// ---------------- input padding (14 -> 16 channels, zero pad) ----------------
__global__ __launch_bounds__(256) void k_pad_input(const float* __restrict__ x,
                                                   float* __restrict__ y, size_t total) {
  size_t i = (size_t)blockIdx.x * 256 + threadIdx.x;
  if (i >= total) return;
  int c = (int)(i & 15);
  size_t m = i >> 4;
  y[i] = (c < 14) ? x[m * 14 + c] : 0.0f;
}

// ---------------- Chebyshev stencil: Xout = scale*(L . Xin) - Xsub ----------------
// Graph: h x w equiangular grid, 4-neighbor, periodic in longitude (w), open in h.
__global__ __launch_bounds__(256) void k_cheb_stencil(
    const float* __restrict__ Xin, const float* __restrict__ Lm,
    int h, int w, int Cp, float scale, const float* __restrict__ Xsub,
    float* __restrict__ Xout, size_t total) {
  size_t i = (size_t)blockIdx.x * 256 + threadIdx.x;
  if (i >= total) return;
  int f = (int)(i % Cp);
  size_t t = i / Cp;
  int V = h * w;
  int v = (int)(t % V);
  int b = (int)(t / V);
  int r = v / w, c = v % w;
  const float* xb = Xin + ((size_t)b * V) * Cp + f;
  const float* Lrow = Lm + (size_t)v * V;
  float s = 0.0f;
  if (r > 0)     { int u = v - w; s += Lrow[u] * xb[(size_t)u * Cp]; }
  if (r < h - 1) { int u = v + w; s += Lrow[u] * xb[(size_t)u * Cp]; }
  { int u = r * w + (c + 1) % w;     s += Lrow[u] * xb[(size_t)u * Cp]; }
  { int u = r * w + (c + w - 1) % w; s += Lrow[u] * xb[(size_t)u * Cp]; }
  float o = scale * s;
  if (Xsub) o -= Xsub[i];
  Xout[i] = o;
}

// ---------------- weight packing into WMMA fragment order ----------------
// Source: (K=3, Fin, Fout). Dest: for global K-row kg (3*Cp rows, zero padded) and
// column n (Wcols, zero padded): Wp[(kg/4)*(4*Wcols) + n*4 + (kg%4)].
// => a lane's B fragment (kidx = 2*laneHi .. 2*laneHi+1) is 2 adjacent floats (b64 load).
__global__ __launch_bounds__(256) void k_pack_w(const float* __restrict__ w, int Fin, int Fout,
                                                int Cp, int Wcols, float* __restrict__ Wp,
                                                size_t total) {
  size_t i = (size_t)blockIdx.x * 256 + threadIdx.x;
  if (i >= total) return;
  int kidx = (int)(i & 3);
  size_t t2 = i >> 2;
  int n = (int)(t2 % Wcols);
  int group = (int)(t2 / Wcols);
  int kg = group * 4 + kidx;
  int kk = kg / Cp;  // cheb term 0..2
  int f = kg % Cp;   // channel within term
  Wp[i] = (f < Fin && n < Fout) ? w[((size_t)kk * Fin + f) * Fout + n] : 0.0f;
}

// ---------------- WMMA GEMM: H = [X0|X1|X2] . W + bias ----------------
// Wave32, one wave = 16x64 output tile (4 x v_wmma_f32_16x16x4_f32 per K-step).
// Fragment layouts per CDNA5 ISA 7.12.2:
//   A:  a[v] = A[m = lane&15][k + 2*(lane>>4) + v]      (one b64 load)
//   B:  b[v] = W[k + 2*(lane>>4) + v][n = lane&15]      (one b64 load, packed layout)
//   C/D VGPR r: row = r + 8*(lane>>4), col = lane&15
__global__ __launch_bounds__(256) void k_cheb_gemm(
    const float* __restrict__ X0, const float* __restrict__ X1,
    const float* __restrict__ X2, int Cp,
    const float* __restrict__ Wp, int Wcols,
    const float* __restrict__ bias, int Fout,
    float* __restrict__ H) {
  const int lane = threadIdx.x & 31;
  const int wave = threadIdx.x >> 5;
  const int laneLo = lane & 15;
  const int laneHi = lane >> 4;
  const int row0 = (blockIdx.x * 8 + wave) * 16;
  const int col0 = blockIdx.y * 64;

  v8f acc[4];
#pragma unroll
  for (int j = 0; j < 4; ++j) acc[j] = (v8f){0.f, 0.f, 0.f, 0.f, 0.f, 0.f, 0.f, 0.f};

  const float* Xs[3] = {X0, X1, X2};
#pragma unroll
  for (int t = 0; t < 3; ++t) {
    const float* Arow = Xs[t] + (size_t)(row0 + laneLo) * Cp + 2 * laneHi;
    // fragment-packed weights: element (kg, n) at Wp[kg*Wcols + n*4 + kg%4] for kg%4==0 base
    const float* Wt = Wp + (size_t)t * Cp * Wcols + ((col0 + laneLo) << 2) + 2 * laneHi;
    for (int k = 0; k < Cp; k += 4) {
      v2f a = *(const v2f*)(Arow + k);            // 8B aligned: Cp%4==0, k%4==0
      const float* Wk = Wt + (size_t)k * Wcols;   // 8B aligned
      v2f b0 = *(const v2f*)(Wk);
      v2f b1 = *(const v2f*)(Wk + 64);
      v2f b2 = *(const v2f*)(Wk + 128);
      v2f b3 = *(const v2f*)(Wk + 192);
      acc[0] = __builtin_amdgcn_wmma_f32_16x16x4_f32(false, a, false, b0, (short)0, acc[0], false, false);
      acc[1] = __builtin_amdgcn_wmma_f32_16x16x4_f32(false, a, false, b1, (short)0, acc[1], false, false);
      acc[2] = __builtin_amdgcn_wmma_f32_16x16x4_f32(false, a, false, b2, (short)0, acc[2], false, false);
      acc[3] = __builtin_amdgcn_wmma_f32_16x16x4_f32(false, a, false, b3, (short)0, acc[3], false, false);
    }
  }

#pragma unroll
  for (int j = 0; j < 4; ++j) {
    int col = col0 + j * 16 + laneLo;
    if (col < Fout) {
      float bv = bias[col];
#pragma unroll
      for (int r = 0; r < 8; ++r) {
        int row = row0 + r + 8 * laneHi;
        H[(size_t)row * Fout + col] = acc[j][r] + bv;
      }
    }
  }
}

// ---------------- BN stats, stage 1: per-chunk per-channel partial sums ----------------
// thread c = tid handles channel c; a wave reads 32 consecutive floats of each row
// (coalesced over row-major H). Deterministic (no atomics).
__global__ __launch_bounds__(256) void k_bn_partial(const float* __restrict__ H, int M, int Fout,
                                                    float* __restrict__ PS) {
  int c = threadIdx.x;
  if (c >= Fout) return;
  int rows = M / NCHUNK;
  const float* p = H + (size_t)blockIdx.x * rows * Fout + c;
  float s = 0.0f, s2 = 0.0f;
  for (int r = 0; r < rows; ++r) {
    float v = p[(size_t)r * Fout];
    s += v;
    s2 += v * v;
  }
  size_t o = (size_t)blockIdx.x * 2 * Fout;
  PS[o + c] = s;
  PS[o + Fout + c] = s2;
}

// ---------------- BN stats, stage 2: reduce chunks, emit mean/biased-var ----------------
__global__ __launch_bounds__(256) void k_bn_final(const float* __restrict__ PS, int M, int Fout,
                                                  float* __restrict__ stats) {
  int c = threadIdx.x;
  if (c >= Fout) return;
  float s = 0.0f, s2 = 0.0f;
  for (int ch = 0; ch < NCHUNK; ++ch) {
    size_t o = (size_t)ch * 2 * Fout;
    s += PS[o + c];
    s2 += PS[o + Fout + c];
  }
  float mean = s / (float)M;
  stats[2 * c] = mean;
  stats[2 * c + 1] = s2 / (float)M - mean * mean;
}

// ---------------- BN apply + ReLU, write padded activation (pad cols zeroed) ----------------
__global__ __launch_bounds__(256) void k_bn_relu(const float* __restrict__ H,
                                                 const float* __restrict__ stats,
                                                 const float* __restrict__ g,
                                                 const float* __restrict__ be,
                                                 int Fout, int Cpout, float* __restrict__ Out,
                                                 size_t total) {
  size_t i = (size_t)blockIdx.x * 256 + threadIdx.x;
  if (i >= total) return;
  int c = (int)(i % Cpout);
  size_t m = i / Cpout;
  float o = 0.0f;
  if (c < Fout) {
    float mean = stats[2 * c], var = stats[2 * c + 1];
    float val = (H[m * Fout + c] - mean) * (g[c] * rsqrtf(var + 1e-5f)) + be[c];
    o = fmaxf(val, 0.0f);
  }
  Out[i] = o;
}

// ---------------- 2x2 max pool with argmax (first-occurrence, q = i2*2 + j2) ----------------
__global__ __launch_bounds__(256) void k_pool2(const float* __restrict__ X, int h, int w,
                                               int C, int Cp, float* __restrict__ Y,
                                               int* __restrict__ idx, size_t total) {
  size_t i = (size_t)blockIdx.x * 256 + threadIdx.x;
  if (i >= total) return;
  int c = (int)(i % C);
  size_t t = i / C;
  int wc2 = w / 2;
  int wcx = (int)(t % wc2); t /= wc2;
  int hcx = (int)(t % (h / 2));
  int b = (int)(t / (h / 2));
  float best = -__builtin_inff();
  int arg = 0;
#pragma unroll
  for (int q = 0; q < 4; ++q) {
    int i2 = q >> 1, j2 = q & 1;
    int v = (2 * hcx + i2) * w + (2 * wcx + j2);
    float val = X[((size_t)b * h * w + v) * Cp + c];
    if (val > best) { best = val; arg = q; }
  }
  size_t o = (size_t)b * (h / 2) * wc2 + (size_t)hcx * wc2 + wcx;
  Y[o * C + c] = best;
  idx[o * C + c] = arg;
}

// ---------------- unpool (scatter to argmax position, zero elsewhere) ----------------
__global__ __launch_bounds__(256) void k_unpool2(const float* __restrict__ X,
                                                 const int* __restrict__ idx, int hc, int wc,
                                                 int C, int Cp, int Cpout, int colOff,
                                                 float* __restrict__ Y, size_t total) {
  size_t i = (size_t)blockIdx.x * 256 + threadIdx.x;
  if (i >= total) return;
  int c = (int)(i % C);
  size_t t = i / C;
  int j = (int)(t % wc); t /= wc;
  int r = (int)(t % hc);
  int b = (int)(t / hc);
  size_t vc = (size_t)b * hc * wc + (size_t)r * wc + j;
  float val = X[vc * Cp + c];
  int a = idx[vc * C + c];
  int wf = 2 * wc;
  size_t Vf = (size_t)4 * hc * wc;
#pragma unroll
  for (int q = 0; q < 4; ++q) {
    int vf = (2 * r + (q >> 1)) * wf + 2 * j + (q & 1);
    Y[((size_t)b * Vf + vf) * Cpout + colOff + c] = (q == a) ? val : 0.0f;
  }
}

// ---------------- channel-slice copy (for skip concat) ----------------
__global__ __launch_bounds__(256) void k_copy_cols(const float* __restrict__ X, int Cs, int C,
                                                   float* __restrict__ Y, int Cd, int off,
                                                   size_t total) {
  size_t i = (size_t)blockIdx.x * 256 + threadIdx.x;
  if (i >= total) return;
  int c = (int)(i % C);
  size_t m = i / C;
  Y[m * Cd + off + c] = X[m * Cs + c];
}

// ============================ host orchestration ============================
extern "C" void kernel_launch(void* const* d_in, const int* in_sizes, int n_in,
                              void* d_out, int out_size, void* d_ws, size_t ws_size,
                              hipStream_t stream) {
  (void)in_sizes; (void)n_in; (void)out_size; (void)ws_size;
  const float* x  = (const float*)d_in[0];
  const float* L0 = (const float*)d_in[1];
  const float* L1 = (const float*)d_in[2];
  const float* L2 = (const float*)d_in[3];

  const int M0 = NB * 64 * 32;  // 262144 rows
  const int M1 = NB * 32 * 16;  // 65536
  const int M2 = NB * 16 * 8;   // 16384

  char* ws = (char*)d_ws;
  size_t off = 0;
  auto alloc = [&](size_t elems) {
    float* p = (float*)(ws + off);
    off += ((elems * 4 + 255) & ~(size_t)255);
    return p;
  };
  float* A0   = alloc((size_t)M0 * 64);
  float* A1   = alloc((size_t)M0 * 32);
  float* E1   = alloc((size_t)M0 * 64);
  float* CAT0 = alloc((size_t)M0 * 128);
  float* P1   = alloc((size_t)M1 * 128);
  float* P2   = alloc((size_t)M1 * 128);
  float* E2   = alloc((size_t)M1 * 128);
  float* CAT1 = alloc((size_t)M1 * 256);
  float* Q1   = alloc((size_t)M2 * 224);
  float* Q2   = alloc((size_t)M2 * 176);
  float* E3   = alloc((size_t)M2 * 256);
  float* X1b  = alloc((size_t)M0 * 128);
  float* X2b  = alloc((size_t)M0 * 128);
  float* Hb   = alloc((size_t)M0 * 64);
  float* WP   = alloc((size_t)768 * 256);
  float* PS   = alloc((size_t)NCHUNK * 2 * 256);
  float* ST   = alloc(512);
  int* IDX1 = (int*)alloc((size_t)M1 * 64);
  int* IDX2 = (int*)alloc((size_t)M2 * 128);

  auto nb = [](size_t n) { return (unsigned)((n + 255) / 256); };

  // one Chebyshev conv block: stencils -> WMMA GEMM(+bias) -> [BN stats -> BN+ReLU]
  auto conv = [&](const float* Xin, int Cin, int Cpin, const float* L, int h, int w,
                  int layer, float* Out, int Cpout, bool bn, int Fout) {
    const float* wgt = (const float*)d_in[4 + 4 * layer + 0];
    const float* bia = (const float*)d_in[4 + 4 * layer + 1];
    const float* gam = (const float*)d_in[4 + 4 * layer + 2];
    const float* bet = (const float*)d_in[4 + 4 * layer + 3];
    const int M = NB * h * w;
    const int Wcols = ((Fout + 63) / 64) * 64;
    size_t npk = (size_t)3 * Cpin * Wcols;
    k_pack_w<<<nb(npk), 256, 0, stream>>>(wgt, Cin, Fout, Cpin, Wcols, WP, npk);
    size_t ne = (size_t)M * Cpin;
    k_cheb_stencil<<<nb(ne), 256, 0, stream>>>(Xin, L, h, w, Cpin, 1.0f, nullptr, X1b, ne);
    k_cheb_stencil<<<nb(ne), 256, 0, stream>>>(X1b, L, h, w, Cpin, 2.0f, Xin, X2b, ne);
    float* Hdst = bn ? Hb : Out;
    dim3 g(M / 128, Wcols / 64);  // 8 waves/block, 16 rows/wave; M is a multiple of 128
    k_cheb_gemm<<<g, 256, 0, stream>>>(Xin, X1b, X2b, Cpin, WP, Wcols, bia, Fout, Hdst);
    if (bn) {
      k_bn_partial<<<NCHUNK, 256, 0, stream>>>(Hb, M, Fout, PS);
      k_bn_final<<<1, 256, 0, stream>>>(PS, M, Fout, ST);
      size_t no = (size_t)M * Cpout;
      k_bn_relu<<<nb(no), 256, 0, stream>>>(Hb, ST, gam, bet, Fout, Cpout, Out, no);
    }
  };

  // ---- encode ----
  size_t np = (size_t)M0 * 16;
  k_pad_input<<<nb(np), 256, 0, stream>>>(x, A0, np);
  conv(A0, 14, 16, L0, 64, 32, 0, A1, 16, true, 16);     // conv11
  conv(A1, 16, 16, L0, 64, 32, 1, A0, 32, true, 32);     // conv12
  conv(A0, 32, 32, L0, 64, 32, 2, E1, 64, true, 64);     // conv13
  k_pool2<<<nb((size_t)M1 * 64), 256, 0, stream>>>(E1, 64, 32, 64, 64, P1, IDX1, (size_t)M1 * 64);
  conv(P1, 64, 64, L1, 32, 16, 3, P2, 88, true, 88);     // conv21
  conv(P2, 88, 88, L1, 32, 16, 4, P1, 112, true, 110);   // conv22
  conv(P1, 110, 112, L1, 32, 16, 5, E2, 128, true, 128); // conv23
  k_pool2<<<nb((size_t)M2 * 128), 256, 0, stream>>>(E2, 32, 16, 128, 128, Q1, IDX2, (size_t)M2 * 128);
  conv(Q1, 128, 128, L2, 16, 8, 6, Q2, 176, true, 174);  // conv31
  conv(Q2, 174, 176, L2, 16, 8, 7, Q1, 220, true, 218);  // conv32
  conv(Q1, 218, 220, L2, 16, 8, 8, E3, 256, true, 256);  // conv33

  // ---- decode ----
  conv(E3, 256, 256, L2, 16, 8, 9, Q2, 128, true, 128);  // convT2
  k_unpool2<<<nb((size_t)M2 * 128), 256, 0, stream>>>(Q2, IDX2, 16, 8, 128, 128, 256, 0, CAT1,
                                                      (size_t)M2 * 128);
  k_copy_cols<<<nb((size_t)M1 * 128), 256, 0, stream>>>(E2, 128, 128, CAT1, 256, 128,
                                                        (size_t)M1 * 128);
  conv(CAT1, 256, 256, L1, 32, 16, 10, P2, 128, true, 128); // uconv21
  conv(P2, 128, 128, L1, 32, 16, 11, P1, 128, true, 128);   // uconv22
  conv(P1, 128, 128, L1, 32, 16, 12, P2, 64, true, 64);     // convT1
  k_unpool2<<<nb((size_t)M1 * 64), 256, 0, stream>>>(P2, IDX1, 32, 16, 64, 64, 128, 0, CAT0,
                                                     (size_t)M1 * 64);
  k_copy_cols<<<nb((size_t)M0 * 64), 256, 0, stream>>>(E1, 64, 64, CAT0, 128, 64,
                                                       (size_t)M0 * 64);
  conv(CAT0, 128, 128, L0, 64, 32, 13, A0, 64, true, 64);   // uconv11
  conv(A0, 64, 64, L0, 64, 32, 14, A1, 32, true, 32);       // uconv12
  conv(A1, 32, 32, L0, 64, 32, 15, (float*)d_out, 7, false, 7); // uconv13 (bias only)
}